// MultiInputConstQuantizationNet_84396107366445
// MI455X (gfx1250) — compile-verified
//
#include <hip/hip_runtime.h>
#include <stdint.h>

#define NCH     117
#define SPATIAL 1296   // 36*36
#define THREADS 256    // 8 wave32s per WGP-resident block

// ---- CDNA5 async global->LDS path (gfx1250), guarded so compile never breaks ----
#if defined(__gfx1250__) && __has_builtin(__builtin_amdgcn_global_load_async_to_lds_b32)
#define HAVE_ASYNC_LDS 1
typedef __attribute__((address_space(1))) int* as1i_t;  // global int*
typedef __attribute__((address_space(3))) int* as3i_t;  // LDS int*
#else
#define HAVE_ASYNC_LDS 0
#endif

__device__ __forceinline__ unsigned int ford(float f) {
  // Monotone float -> uint mapping (no NaNs in this workload; no -0 sources).
  unsigned int u = __float_as_uint(f);
  return (u & 0x80000000u) ? ~u : (u | 0x80000000u);
}

// Every element of the (117,36,36) result maps to exactly one global source.
__device__ __forceinline__ const float* src_of(
    int c, int i,
    const float* __restrict__ x,      // (16,32,32)
    const float* __restrict__ cat1,   // (16,1,32)
    const float* __restrict__ cat2,   // (16,3,32)
    const float* __restrict__ ccat,   // (16,36,4)
    const float* __restrict__ cc1,    // (1,36,36)
    const float* __restrict__ cc2,    // (2,36,36)
    const float* __restrict__ cc3,    // (3,36,36)
    const float* __restrict__ st1,    // (39,36,36)
    const float* __restrict__ st2) {  // (39,36,36)
  if (c < 39)  return st1 + c * SPATIAL + i;
  if (c >= 78) return st2 + (c - 78) * SPATIAL + i;
  int cc = c - 39;                       // channel inside the computed 39-ch block
  if (cc == 0 || cc == 38)  return cc1 + i;
  if (cc >= 17 && cc <= 18) return cc2 + (cc - 17) * SPATIAL + i;
  if (cc >= 35 && cc <= 37) return cc3 + (cc - 35) * SPATIAL + i;
  int k = (cc <= 16) ? (cc - 1) : (cc - 19);  // x36 channel 0..15
  int r = i / 36;
  int col = i - r * 36;
  if (col < 4) return ccat + (k * 36 + r) * 4 + col;     // axis-3 prefix
  int c2 = col - 4;
  if (r == 0)  return cat1 + k * 32 + c2;                // axis-2 prefix row
  if (r <= 32) return x + (k * 32 + (r - 1)) * 32 + c2;  // payload
  return cat2 + (k * 3 + (r - 33)) * 32 + c2;            // axis-2 suffix rows
}

__global__ __launch_bounds__(THREADS)
void fused_concat_argmax_add(
    const float* __restrict__ x,
    const float* __restrict__ cat1,
    const float* __restrict__ cat2,
    const float* __restrict__ ccat,
    const float* __restrict__ cc1,
    const float* __restrict__ cc2,
    const float* __restrict__ cc3,
    const float* __restrict__ st1,
    const float* __restrict__ st2,
    const float* __restrict__ gather_const,   // (2592,)
    float* __restrict__ out) {                // (117,36,36)
  __shared__ float vals[SPATIAL];
  __shared__ unsigned long long skey;
  __shared__ float sb;

  const int c   = blockIdx.x;
  const int tid = threadIdx.x;

  // ---- Phase A: gather this channel into LDS (async DMA path on gfx1250) ----
  for (int i = tid; i < SPATIAL; i += THREADS) {
    const float* s = src_of(c, i, x, cat1, cat2, ccat, cc1, cc2, cc3, st1, st2);
#if HAVE_ASYNC_LDS
    __builtin_amdgcn_global_load_async_to_lds_b32(
        (as1i_t)(float*)s, (as3i_t)(float*)&vals[i],
        /*offset=*/0, /*cpol=*/0);
#else
    vals[i] = *s;
#endif
  }
  if (tid == 0) skey = 0ull;
#if HAVE_ASYNC_LDS
#if __has_builtin(__builtin_amdgcn_s_wait_asynccnt)
  __builtin_amdgcn_s_wait_asynccnt(0);
#else
  asm volatile("s_wait_asynccnt 0" ::: "memory");
#endif
#endif
  __syncthreads();   // all waves drained their async copies before signaling

  // ---- Phase B: argmax with first-index tie-break (exact jnp.argmax) ----
  unsigned long long best = 0ull;
  for (int i = tid; i < SPATIAL; i += THREADS) {
    unsigned long long key =
        ((unsigned long long)ford(vals[i]) << 32) |
        (unsigned long long)(0xFFFFFFFFu - (unsigned)i);
    if (key > best) best = key;
  }
  // wave32 butterfly reduction, then one ds_max_u64 per wave
  for (int off = 16; off > 0; off >>= 1) {
    unsigned long long o = __shfl_xor(best, off, 32);
    if (o > best) best = o;
  }
  if ((tid & 31) == 0) atomicMax(&skey, best);
  __syncthreads();

  if (tid == 0) {
    unsigned idx = 0xFFFFFFFFu - (unsigned)(skey & 0xFFFFFFFFull);
    sb = gather_const[idx];
  }
  __syncthreads();

  // ---- Phase C: broadcast-add and coalesced store ----
  const float b = sb;
  float* o = out + c * SPATIAL;
  for (int i = tid; i < SPATIAL; i += THREADS) {
    o[i] = vals[i] + b;
  }
}

extern "C" void kernel_launch(void* const* d_in, const int* in_sizes, int n_in,
                              void* d_out, int out_size, void* d_ws, size_t ws_size,
                              hipStream_t stream) {
  const float* x    = (const float*)d_in[0];  // (1,16,32,32)
  const float* cat1 = (const float*)d_in[1];  // (1,16,1,32)
  const float* cat2 = (const float*)d_in[2];  // (1,16,3,32)
  const float* ccat = (const float*)d_in[3];  // (1,16,36,4)
  const float* cc1  = (const float*)d_in[4];  // (1,1,36,36)
  const float* cc2  = (const float*)d_in[5];  // (1,2,36,36)
  const float* cc3  = (const float*)d_in[6];  // (1,3,36,36)
  const float* st1  = (const float*)d_in[7];  // (1,39,36,36)
  const float* st2  = (const float*)d_in[8];  // (1,39,36,36)
  const float* gth  = (const float*)d_in[9];  // (1,2592)
  float* out = (float*)d_out;                 // (1,117,36,36)
  (void)in_sizes; (void)n_in; (void)out_size; (void)d_ws; (void)ws_size;

  fused_concat_argmax_add<<<NCH, THREADS, 0, stream>>>(
      x, cat1, cat2, ccat, cc1, cc2, cc3, st1, st2, gth, out);
}